// GraphEncoder_1623497638364
// MI455X (gfx1250) — compile-verified
//
#include <hip/hip_runtime.h>
#include <hip/hip_bf16.h>

typedef float v2f __attribute__((ext_vector_type(2)));
typedef float v8f __attribute__((ext_vector_type(8)));

#define D 64   // feature dim (D_IN == D_HID == 64)

// ---------------------------------------------------------------------------
// Degree / normalization
// ---------------------------------------------------------------------------
__global__ void k_deg_init(float* __restrict__ deg, int n) {
  int i = blockIdx.x * blockDim.x + threadIdx.x;
  if (i < n) deg[i] = 1.0f;              // self-loop contributes 1 to every node
}

__global__ void k_deg_accum(const int* __restrict__ dst, float* __restrict__ deg, int E) {
  int e = blockIdx.x * blockDim.x + threadIdx.x;
  if (e < E) atomicAdd(&deg[dst[e]], 1.0f);
}

__global__ void k_dinv(float* __restrict__ deg, int n) {
  int i = blockIdx.x * blockDim.x + threadIdx.x;
  if (i < n) {
    float d = deg[i];
    deg[i] = (d > 0.0f) ? rsqrtf(d) : 0.0f;
  }
}

__global__ void k_norm(const int* __restrict__ src, const int* __restrict__ dst,
                       const float* __restrict__ dinv, float* __restrict__ norm,
                       int E, int n) {
  int e = blockIdx.x * blockDim.x + threadIdx.x;
  int tot = E + n;
  if (e < tot) {
    int s, t;
    if (e < E) { s = src[e]; t = dst[e]; }
    else       { s = t = e - E; }        // appended self-loops
    norm[e] = dinv[s] * dinv[t];
  }
}

// ---------------------------------------------------------------------------
// Dense GEMM  C[M x 64] = A[M x 64] * W[64 x 64]  via V_WMMA_F32_16X16X4_F32.
// One wave computes a 16-row tile across all four 16-col N-tiles.
// VGPR layouts per CDNA5 ISA 7.12.2:
//   A 16x4 f32 : lanes 0-15 row=lane, VGPR{0,1}=K{0,1}; lanes 16-31 K{2,3}
//   B  4x16 f32: VGPR0 = row K=0 (low lanes) / K=2 (high); VGPR1 = K=1 / K=3
//   C/D 16x16  : VGPR v -> M=v (low lanes), M=v+8 (high lanes), N = lane%16
//
// W is staged into LDS pre-paired along K (float2 {W[2kk][n], W[2kk+1][n]})
// so every B fragment is one contiguous ds_load_b64 into an even VGPR pair.
// A rows are clamped (not predicated) so the A stream is a clean
// global_load_b64 per k-step with no exec-mask branching; the D-matrix
// stores carry the real row guard.
// ---------------------------------------------------------------------------
__global__ void k_gemm_wmma(const float* __restrict__ A, const float* __restrict__ W,
                            float* __restrict__ Cout, int M) {
  __shared__ v2f sW[(D / 2) * D];        // 16 KB: sW[kk*D + n] = {W[2kk][n], W[2kk+1][n]}
  for (int i = threadIdx.x; i < (D / 2) * D; i += blockDim.x) {
    int kk = i / D;
    int nn = i - kk * D;
    v2f p;
    p.x = W[(2 * kk + 0) * D + nn];
    p.y = W[(2 * kk + 1) * D + nn];
    sW[i] = p;
  }
  __syncthreads();

  int lane = threadIdx.x & 31;
  int half = lane >> 4;                  // 0: lanes 0-15, 1: lanes 16-31
  int lid  = lane & 15;
  int wave = blockIdx.x * (blockDim.x >> 5) + (threadIdx.x >> 5);
  int row0 = wave * 16;
  if (row0 >= M) return;                 // wave-uniform: EXEC stays all-ones

  v8f acc0 = {}, acc1 = {}, acc2 = {}, acc3 = {};

  int r  = row0 + lid;
  int rc = (r < M) ? r : (M - 1);        // clamp instead of predicate
  const v2f* arow = (const v2f*)(A + (size_t)rc * D);

#pragma unroll
  for (int k = 0; k < D; k += 4) {
    int kk = (k >> 1) + half;            // pair index for this lane-half (K = 2kk, 2kk+1)
    v2f a = arow[kk];                    // one global_load_b64
    const v2f* wrow = &sW[kk * D + lid];
    v2f b0 = wrow[0];                    // four ds_load_b64
    v2f b1 = wrow[16];
    v2f b2 = wrow[32];
    v2f b3 = wrow[48];
    acc0 = __builtin_amdgcn_wmma_f32_16x16x4_f32(false, a, false, b0, (short)0, acc0, false, false);
    acc1 = __builtin_amdgcn_wmma_f32_16x16x4_f32(false, a, false, b1, (short)0, acc1, false, false);
    acc2 = __builtin_amdgcn_wmma_f32_16x16x4_f32(false, a, false, b2, (short)0, acc2, false, false);
    acc3 = __builtin_amdgcn_wmma_f32_16x16x4_f32(false, a, false, b3, (short)0, acc3, false, false);
  }

#pragma unroll
  for (int v = 0; v < 8; ++v) {
    int rr = row0 + v + 8 * half;
    if (rr < M) {
      float* crow = Cout + (size_t)rr * D + lid;
      crow[0]  = acc0[v];
      crow[16] = acc1[v];
      crow[32] = acc2[v];
      crow[48] = acc3[v];
    }
  }
}

// ---------------------------------------------------------------------------
// Aggregation buffer init with bias, edge scatter-add, PReLU
// ---------------------------------------------------------------------------
__global__ void k_bias_init(float* __restrict__ agg, const float* __restrict__ b, int n) {
  int i = blockIdx.x * blockDim.x + threadIdx.x;
  if (i < n * D) agg[i] = b[i & (D - 1)];
}

__global__ void k_scatter(const int* __restrict__ src, const int* __restrict__ dst,
                          const float* __restrict__ norm, const float* __restrict__ h,
                          float* __restrict__ agg, int E, int n) {
  int tid = blockIdx.x * blockDim.x + threadIdx.x;
  int d = tid & (D - 1);
  int e = tid >> 6;
  int tot = E + n;
  if (e < tot) {
    int s, t;
    if (e < E) { s = src[e]; t = dst[e]; }
    else       { s = t = e - E; }
    float v = h[(size_t)s * D + d] * norm[e];
    atomicAdd(&agg[(size_t)t * D + d], v);
  }
}

__global__ void k_prelu(const float* __restrict__ in, const float* __restrict__ a,
                        float* __restrict__ out, int n) {
  int i = blockIdx.x * blockDim.x + threadIdx.x;
  if (i < n * D) {
    float v = in[i];
    float al = a[i & (D - 1)];
    out[i] = (v >= 0.0f) ? v : al * v;
  }
}

// ---------------------------------------------------------------------------
// Launch
// ---------------------------------------------------------------------------
static inline int cdiv_ll(long long a, long long b) { return (int)((a + b - 1) / b); }

extern "C" void kernel_launch(void* const* d_in, const int* in_sizes, int n_in,
                              void* d_out, int out_size, void* d_ws, size_t ws_size,
                              hipStream_t stream) {
  (void)n_in; (void)out_size; (void)ws_size;

  const float* x  = (const float*)d_in[0];
  const int*   ei = (const int*)  d_in[1];   // edge_index [2, E] flattened
  const float* W1 = (const float*)d_in[2];
  const float* b1 = (const float*)d_in[3];
  const float* a1 = (const float*)d_in[4];
  const float* W2 = (const float*)d_in[5];
  const float* b2 = (const float*)d_in[6];
  const float* a2 = (const float*)d_in[7];

  const int n = in_sizes[0] / D;
  const int E = in_sizes[1] / 2;
  const int* src = ei;
  const int* dst = ei + E;

  // Workspace carve-out (floats): dinv[n] | norm[E+n] | bufA[n*D] | bufB[n*D]
  float* ws   = (float*)d_ws;
  float* dinv = ws;
  float* norm = dinv + n;
  float* bufA = norm + (size_t)(E + n);
  float* bufB = bufA + (size_t)n * D;
  float* out  = (float*)d_out;

  const int B = 256;
  const int gN    = cdiv_ll(n, B);
  const int gE    = cdiv_ll(E, B);
  const int gEN   = cdiv_ll((long long)E + n, B);
  const int gND   = cdiv_ll((long long)n * D, B);
  const int gSc   = cdiv_ll(((long long)E + n) * D, B);
  const int gGemm = cdiv_ll((long long)cdiv_ll(n, 16) * 32, B);

  // --- normalization (shared by both layers) ---
  k_deg_init <<<gN,  B, 0, stream>>>(dinv, n);
  k_deg_accum<<<gE,  B, 0, stream>>>(dst, dinv, E);
  k_dinv     <<<gN,  B, 0, stream>>>(dinv, n);
  k_norm     <<<gEN, B, 0, stream>>>(src, dst, dinv, norm, E, n);

  // --- layer 1 ---
  k_gemm_wmma<<<gGemm, B, 0, stream>>>(x, W1, bufA, n);
  k_bias_init<<<gND,   B, 0, stream>>>(bufB, b1, n);
  k_scatter  <<<gSc,   B, 0, stream>>>(src, dst, norm, bufA, bufB, E, n);
  k_prelu    <<<gND,   B, 0, stream>>>(bufB, a1, bufA, n);

  // --- layer 2 ---
  k_gemm_wmma<<<gGemm, B, 0, stream>>>(bufA, W2, bufB, n);
  k_bias_init<<<gND,   B, 0, stream>>>(out, b2, n);
  k_scatter  <<<gSc,   B, 0, stream>>>(src, dst, norm, bufB, out, E, n);
  k_prelu    <<<gND,   B, 0, stream>>>(out, a2, out, n);
}